// CenterAlignment_86199993630993
// MI455X (gfx1250) — compile-verified
//
#include <hip/hip_runtime.h>
#include <hip/hip_bf16.h>

#define NUM_CLASSES 1000
#define FEA_DIM     1024
#define BATCH       32768
#define MOMENTUM    0.9f

typedef __attribute__((ext_vector_type(2))) float v2f;
typedef __attribute__((ext_vector_type(4))) float v4f;
typedef __attribute__((ext_vector_type(8))) float v8f;

// workspace layout in floats
#define WS_SUMS   0
#define WS_COUNTS (NUM_CLASSES * FEA_DIM)
#define WS_LOSS   (WS_COUNTS + NUM_CLASSES)
#define WS_NPRES  (WS_LOSS + 1)
#define WS_TOTAL  (WS_NPRES + 1)

__global__ void zero_ws_kernel(float* __restrict__ ws) {
  int i = blockIdx.x * blockDim.x + threadIdx.x;
  int stride = gridDim.x * blockDim.x;
  for (int j = i; j < WS_TOTAL; j += stride) ws[j] = 0.0f;
}

// One block per label pair (rows i and i+BATCH share a label).
// 256 threads cover 1024 floats as float4 chunks; merge the two crops
// before the atomic to halve atomic traffic (32M atomics total).
// x is a 256MB single-use stream: load it NON-TEMPORAL so it does not
// churn the 192MB L2 and evict the 4MB sums tile the atomics live in.
// Thread 0 also bumps the per-class count (+2: both crops).
__global__ void scatter_kernel(const float* __restrict__ x,
                               const int* __restrict__ l,
                               float* __restrict__ sums,
                               float* __restrict__ counts) {
  int pair = blockIdx.x;      // 0..BATCH-1
  int d4   = threadIdx.x;     // 0..255 -> float4 chunk
  int c    = l[pair];         // uniform -> scalar load
  if (d4 == 0) {
    unsafeAtomicAdd(&counts[c], 2.0f);
  }
  const v4f* r0 = (const v4f*)(x + (size_t)pair * FEA_DIM);
  const v4f* r1 = (const v4f*)(x + (size_t)(pair + BATCH) * FEA_DIM);
  v4f a = __builtin_nontemporal_load(r0 + d4);   // global_load_b128 th:NT
  v4f b = __builtin_nontemporal_load(r1 + d4);
  float* dst = sums + (size_t)c * FEA_DIM + (size_t)d4 * 4;
  unsafeAtomicAdd(dst + 0, a.x + b.x);
  unsafeAtomicAdd(dst + 1, a.y + b.y);
  unsafeAtomicAdd(dst + 2, a.z + b.z);
  unsafeAtomicAdd(dst + 3, a.w + b.w);
}

// One wave (32 lanes) per 16-class tile. Computes diagonals of
// upd*upd^T, upd*cs^T, cs*cs^T via V_WMMA_F32_16X16X4_F32 accumulated
// over FEA_DIM/4 = 256 k-steps, then:
//   per_cls = || upd/||upd|| - cs ||^2 = 1 - 2*(upd.cs)/||upd|| + ||cs||^2
// A-operand layout for 16x4 f32: lane L (0-15): M=L, K={0,1}; lane L
// (16-31): M=L-16, K={2,3}. The transposed B-operand (4x16) has the
// identical per-lane striping, so the same registers serve A and B.
__global__ void finalize_kernel(const float* __restrict__ ci,
                                const float* __restrict__ cs,
                                float* __restrict__ ws) {
  __shared__ float sg[3][32][8];
  __shared__ float redp[32];
  __shared__ float redn[32];

  const float* sums   = ws + WS_SUMS;
  const float* counts = ws + WS_COUNTS;

  int lane = threadIdx.x;                 // 0..31, single wave
  int c    = blockIdx.x * 16 + (lane & 15);
  int cc   = (c < NUM_CLASSES) ? c : (NUM_CLASSES - 1);  // clamp, keep EXEC full
  float cnt   = counts[cc];
  float inv01 = (1.0f - MOMENTUM) / fmaxf(cnt, 1.0f);

  const float* ci_row = ci   + (size_t)cc * FEA_DIM;
  const float* cs_row = cs   + (size_t)cc * FEA_DIM;
  const float* su_row = sums + (size_t)cc * FEA_DIM;
  int khalf = (lane >> 4) * 2;            // 0 for lanes 0-15, 2 for 16-31

  v8f g1 = {}; v8f g2 = {}; v8f g3 = {};
  for (int kk = 0; kk < FEA_DIM / 4; ++kk) {
    int k = kk * 4 + khalf;
    float2 u = *(const float2*)(ci_row + k);
    float2 s = *(const float2*)(su_row + k);
    float2 t = *(const float2*)(cs_row + k);
    v2f a, b;
    a.x = u.x * MOMENTUM + s.x * inv01;   // upd = 0.9*ci + 0.1*mean
    a.y = u.y * MOMENTUM + s.y * inv01;
    b.x = t.x;
    b.y = t.y;
    g1 = __builtin_amdgcn_wmma_f32_16x16x4_f32(false, a, false, a, (short)0, g1, false, false);
    g2 = __builtin_amdgcn_wmma_f32_16x16x4_f32(false, a, false, b, (short)0, g2, false, false);
    g3 = __builtin_amdgcn_wmma_f32_16x16x4_f32(false, b, false, b, (short)0, g3, false, false);
  }

#pragma unroll
  for (int r = 0; r < 8; ++r) {
    sg[0][lane][r] = g1[r];
    sg[1][lane][r] = g2[r];
    sg[2][lane][r] = g3[r];
  }
  __syncthreads();

  // diagonal (m,m): m<8 -> VGPR m of lane m; m>=8 -> VGPR m-8 of lane m+16
  int m       = lane & 15;
  int srcLane = (m < 8) ? m : (m + 16);
  int r       = (m < 8) ? m : (m - 8);
  float a1 = sg[0][srcLane][r];   // ||upd||^2
  float a2 = sg[1][srcLane][r];   // upd . cs
  float a3 = sg[2][srcLane][r];   // ||cs||^2

  float per = 0.0f, pres = 0.0f;
  bool active = (lane < 16) && (c < NUM_CLASSES) && (cnt > 0.0f);
  if (active) {
    per  = 1.0f - 2.0f * a2 / sqrtf(a1) + a3;
    pres = 1.0f;
  }
  redp[lane] = per;
  redn[lane] = pres;
  __syncthreads();
  if (lane == 0) {
    float sp = 0.0f, sn = 0.0f;
    for (int i = 0; i < 16; ++i) { sp += redp[i]; sn += redn[i]; }
    unsafeAtomicAdd(ws + WS_LOSS,  sp);
    unsafeAtomicAdd(ws + WS_NPRES, sn);
  }
}

__global__ void final_div_kernel(const float* __restrict__ ws, float* __restrict__ out) {
  if (threadIdx.x == 0 && blockIdx.x == 0) {
    out[0] = ws[WS_LOSS] / ws[WS_NPRES];
  }
}

extern "C" void kernel_launch(void* const* d_in, const int* in_sizes, int n_in,
                              void* d_out, int out_size, void* d_ws, size_t ws_size,
                              hipStream_t stream) {
  const float* x  = (const float*)d_in[0];   // [2B, 1024]
  const float* ci = (const float*)d_in[1];   // [1000, 1024]
  const float* cs = (const float*)d_in[2];   // [1000, 1024]
  const int*   l  = (const int*)d_in[3];     // [B]
  float* ws  = (float*)d_ws;
  float* out = (float*)d_out;

  zero_ws_kernel<<<512, 256, 0, stream>>>(ws);
  scatter_kernel<<<BATCH, FEA_DIM / 4, 0, stream>>>(x, l, ws + WS_SUMS, ws + WS_COUNTS);
  finalize_kernel<<<(NUM_CLASSES + 15) / 16, 32, 0, stream>>>(ci, cs, ws);
  final_div_kernel<<<1, 1, 0, stream>>>(ws, out);
}